// SpectralLoss_33956011442448
// MI455X (gfx1250) — compile-verified
//
#include <hip/hip_runtime.h>
#include <hip/hip_bf16.h>

typedef float v2f __attribute__((ext_vector_type(2)));
typedef float v8f __attribute__((ext_vector_type(8)));

#define W_IMG 512
#define H_IMG 512
#define NBLOCKS 512

// One wave processes 16x16 super-tiles (= four 8x8 DCT blocks) with
// V_WMMA_F32_16X16X4_F32 using a block-diagonal diag(T,T) operand.
__global__ __launch_bounds__(256) void spectral_loss_stage1(
    const float* __restrict__ in, const float* __restrict__ tg,
    const float* __restrict__ T, const float* __restrict__ Wq,
    float* __restrict__ partials, int nTiles)
{
    __shared__ float lds_t[8][16 * 18];   // per-wave transpose pad-18 buffer
    __shared__ float red[256];

    const int lane = threadIdx.x & 31;
    const int wave = threadIdx.x >> 5;
    float* tp = &lds_t[wave][0];

    const int r  = lane & 15;           // operand row index
    const int cb = (lane >> 4) * 2;     // K/column base: 0 or 2 (upper lanes)

    // A-operand slices of block-diagonal T_bd (16x16), lane-resident.
    // A 16x4 layout: lane L holds row (L%16); VGPR0 = K = 4k+cb, VGPR1 = K+1.
    v2f ta[4];
#pragma unroll
    for (int k = 0; k < 4; ++k) {
        const int c0 = 4 * k + cb;
        float x0 = 0.f, x1 = 0.f;
        if ((r < 8) == (c0 < 8))       x0 = T[(r & 7) * 8 + (c0 & 7)];
        if ((r < 8) == ((c0 + 1) < 8)) x1 = T[(r & 7) * 8 + ((c0 + 1) & 7)];
        ta[k][0] = x0; ta[k][1] = x1;
    }

    // Accumulator row j (lane L) is D[j + 8*(L/16)][L%16] -> weight[j][L%8].
    float wrow[8];
#pragma unroll
    for (int j = 0; j < 8; ++j) wrow[j] = Wq[j * 8 + (lane & 7)];

    float local = 0.f;

    const int gwave  = (int)((blockIdx.x * blockDim.x + threadIdx.x) >> 5);
    const int nwaves = (int)((gridDim.x * blockDim.x) >> 5);

    for (int t = gwave; t < nTiles; t += nwaves) {
        const int img = t >> 10;            // 1024 super-tiles per 512x512 image
        const int s   = t & 1023;
        const int th  = s >> 5;
        const int tw  = s & 31;
        const long base = (long)img * (H_IMG * W_IMG) + (long)th * 16 * W_IMG + tw * 16;
        const float* ip = in + base + r * W_IMG;
        const float* gp = tg + base + r * W_IMG;

        // Stage-1 B operand is M^T: element B[4k+cb+j][r] = M[r][4k+cb+j],
        // i.e. contiguous float2 pairs from row r of the tile.
        v2f mb[4];
#pragma unroll
        for (int k = 0; k < 4; ++k) {
            const int c0 = 4 * k + cb;
            v2f a = *(const v2f*)(ip + c0);
            v2f b = *(const v2f*)(gp + c0);
            mb[k] = a - b;
        }

        // P = T_bd * M^T  (16x16x16 as 4 chained 16x16x4 WMMAs)
        v8f p = {};
#pragma unroll
        for (int k = 0; k < 4; ++k)
            p = __builtin_amdgcn_wmma_f32_16x16x4_f32(
                    false, ta[k], false, mb[k], (short)0, p, false, false);

        // Transpose P through LDS (stride 18 -> conflict free).
        // D layout: VGPR j at lane L holds P[j + 8*(L/16)][L%16].
        const int mh = (lane >> 4) * 8;
#pragma unroll
        for (int j = 0; j < 8; ++j)
            tp[(j + mh) * 18 + r] = p[j];

        __builtin_amdgcn_wave_barrier();   // keep DS store->load program order

        // D = T_bd * P^T = T_bd * M * T_bd^T ; B element = P[r][4k+cb+j]
        v8f d = {};
#pragma unroll
        for (int k = 0; k < 4; ++k) {
            const int c0 = 4 * k + cb;
            v2f qb = *(const v2f*)&tp[r * 18 + c0];
            d = __builtin_amdgcn_wmma_f32_16x16x4_f32(
                    false, ta[k], false, qb, (short)0, d, false, false);
        }

        __builtin_amdgcn_wave_barrier();   // loads done before next iter stores

#pragma unroll
        for (int j = 0; j < 8; ++j)
            local = fmaf(d[j] * d[j], wrow[j], local);
    }

    // Deterministic block reduction -> one partial per block.
    red[threadIdx.x] = local;
    __syncthreads();
    for (int off = 128; off > 0; off >>= 1) {
        if ((int)threadIdx.x < off) red[threadIdx.x] += red[threadIdx.x + off];
        __syncthreads();
    }
    if (threadIdx.x == 0) partials[blockIdx.x] = red[0];
}

__global__ __launch_bounds__(256) void spectral_loss_stage2(
    const float* __restrict__ partials, int n, float* __restrict__ out, float inv_cnt)
{
    __shared__ float red[256];
    float s = 0.f;
    for (int i = threadIdx.x; i < n; i += 256) s += partials[i];
    red[threadIdx.x] = s;
    __syncthreads();
    for (int off = 128; off > 0; off >>= 1) {
        if ((int)threadIdx.x < off) red[threadIdx.x] += red[threadIdx.x + off];
        __syncthreads();
    }
    if (threadIdx.x == 0) out[0] = red[0] * inv_cnt;
}

extern "C" void kernel_launch(void* const* d_in, const int* in_sizes, int n_in,
                              void* d_out, int out_size, void* d_ws, size_t ws_size,
                              hipStream_t stream) {
    const float* in = (const float*)d_in[0];   // (16,3,512,512) f32
    const float* tg = (const float*)d_in[1];   // (16,3,512,512) f32
    const float* T  = (const float*)d_in[2];   // (8,8) DCT
    const float* Wq = (const float*)d_in[3];   // (8,8) perceptual weight
    float* out      = (float*)d_out;
    float* partials = (float*)d_ws;

    const long n    = (long)in_sizes[0];       // 12,582,912 elements
    const int nImg  = (int)(n >> 18);          // / (512*512) = 48 images
    const int nTile = nImg * 1024;             // 16x16 super-tiles

    spectral_loss_stage1<<<NBLOCKS, 256, 0, stream>>>(in, tg, T, Wq, partials, nTile);
    spectral_loss_stage2<<<1, 256, 0, stream>>>(partials, NBLOCKS, out, 1.0f / (float)n);
}